// Encoder_71176198029902
// MI455X (gfx1250) — compile-verified
//
#include <hip/hip_runtime.h>
#include <hip/hip_bf16.h>
#include <stdint.h>

// ---------------------------------------------------------------------------
// TAGConv (K=2) x2 + per-graph max readout for MI455X (gfx1250, wave32).
//  - GEMMs: V_WMMA_F32_16X16X4_F32, one wave per 16-row stripe computing ALL
//    output columns (8 accumulators), W K-slabs staged in LDS by the Tensor
//    Data Mover (tensor_load_to_lds, double-buffered, s_wait_tensorcnt).
//  - Propagation: GLOBAL_ATOMIC_ADD_F32 scatter, float4-vectorized.
//  - Readout: uint-bitpattern atomicMax (valid: post-ReLU values >= 0).
// ---------------------------------------------------------------------------

#define N_NODES 50000
#define N_EDGES 600000
#define N_GRAPHS 100
#define IN_DIM 128
#define HID 128
#define OUT_DIM 64
#define CAT 384            // (K+1)*128
#define ROW_TILES (N_NODES / 16)   // 3125, exact

typedef __attribute__((ext_vector_type(2))) float v2f;
typedef __attribute__((ext_vector_type(8))) float v8f;
typedef __attribute__((ext_vector_type(4))) unsigned int v4u;
typedef __attribute__((ext_vector_type(8))) int v8i;
typedef __attribute__((ext_vector_type(4))) int v4i;

#if __has_builtin(__builtin_amdgcn_tensor_load_to_lds) && \
    __has_builtin(__builtin_amdgcn_s_wait_tensorcnt)
#define USE_TDM 1
#else
#define USE_TDM 0
#endif

// ------------------------------- utilities ---------------------------------

__global__ void k_zero_f32(float* __restrict__ p, int n) {
    int i = blockIdx.x * blockDim.x + threadIdx.x;
    if (i < n) p[i] = 0.0f;
}

__global__ void k_degree(const int* __restrict__ dst, float* __restrict__ deg, int nE) {
    int e = blockIdx.x * blockDim.x + threadIdx.x;
    if (e < nE) atomicAdd(&deg[dst[e]], 1.0f);
}

__global__ void k_norm(const float* __restrict__ deg, float* __restrict__ norm, int n) {
    int i = blockIdx.x * blockDim.x + threadIdx.x;
    if (i < n) norm[i] = 1.0f / sqrtf(fmaxf(deg[i], 1.0f));
}

// copy a 128-wide column block: dst[n*ldd + dOff .. +128) = src[n*lds + sOff ..)
__global__ void k_copy128(float* __restrict__ dstp, int ldd, int dOff,
                          const float* __restrict__ srcp, int lds, int sOff) {
    int tid = blockIdx.x * blockDim.x + threadIdx.x;     // N_NODES*32 threads
    int n = tid >> 5;
    if (n >= N_NODES) return;
    int q = (tid & 31) << 2;
    const float4 v = *(const float4*)(srcp + (long)n * lds + sOff + q);
    *(float4*)(dstp + (long)n * ldd + dOff + q) = v;
}

// ----------------------------- propagation ---------------------------------

// agg[dst] += feats[src, srcCol:srcCol+128] * norm[src]   (32 lanes/edge, f4)
__global__ void k_scatter(const float* __restrict__ feats, int srcCol,
                          float* __restrict__ agg,
                          const float* __restrict__ norm,
                          const int* __restrict__ srcI,
                          const int* __restrict__ dstI) {
    long tid = (long)blockIdx.x * blockDim.x + threadIdx.x;
    int e = (int)(tid >> 5);
    if (e >= N_EDGES) return;
    int q = ((int)tid & 31) << 2;
    int s = srcI[e];
    int d = dstI[e];
    float ns = norm[s];
    const float4 v = *(const float4*)(feats + (long)s * CAT + srcCol + q);
    float* dp = agg + (long)d * 128 + q;
    atomicAdd(dp + 0, v.x * ns);
    atomicAdd(dp + 1, v.y * ns);
    atomicAdd(dp + 2, v.z * ns);
    atomicAdd(dp + 3, v.w * ns);
}

// feats[n, dstCol:+128] = agg[n,:] * norm[n]
__global__ void k_scale_store(float* __restrict__ feats, int dstCol,
                              const float* __restrict__ agg,
                              const float* __restrict__ norm) {
    int tid = blockIdx.x * blockDim.x + threadIdx.x;     // N_NODES*32 threads
    int n = tid >> 5;
    if (n >= N_NODES) return;
    int q = (tid & 31) << 2;
    float nn = norm[n];
    float4 v = *(const float4*)(agg + (long)n * 128 + q);
    v.x *= nn; v.y *= nn; v.z *= nn; v.w *= nn;
    *(float4*)(feats + (long)n * CAT + dstCol + q) = v;
}

// ------------------------------ TDM helper ---------------------------------
// 1-D TDM load of `nElems` f32 (contiguous) from gptr into LDS at `ldsOff`.
// D# per CDNA5 ISA: group0 {count=1 | lds_addr |
// global_addr[56:0] | type=2}, group1 {data_size=4B, tensor_dim0/1,
// tile_dim0/1}, groups 2/3 (and the extra high-dim words) zero for <=2D.
#if USE_TDM
__device__ __forceinline__ void tdm_load_1d(const float* gptr, unsigned int ldsOff,
                                            int nElems, int remElems) {
    unsigned long long ga = (unsigned long long)(uintptr_t)gptr;
    v4u g0;
    g0.x = 1u;                                           // count=1 (valid), user mode
    g0.y = ldsOff;                                       // lds_addr (bytes)
    g0.z = (unsigned int)(ga & 0xFFFFFFFFull);           // global_addr[31:0]
    g0.w = (unsigned int)((ga >> 32) & 0x01FFFFFFull)    // global_addr[56:32]
         | (2u << 30);                                   // type=2 ("image")
    v8i g1;
    g1[0] = 0x00020000;                                  // data_size=2 (4 bytes)
    g1[1] = (int)(((unsigned)remElems & 0xFFFFu) << 16); // tensor_dim0[15:0] @ bits63:48
    g1[2] = (int)((((unsigned)remElems >> 16) & 0xFFFFu) // tensor_dim0[31:16]
         | (1u << 16));                                  // tensor_dim1 = 1
    g1[3] = (int)(((unsigned)nElems & 0xFFFFu) << 16);   // tile_dim0 @ bits127:112
    g1[4] = 1;                                           // tile_dim1 = 1, tile_dim2 = 0
    g1[5] = nElems;                                      // tensor_dim0_stride[31:0]
    g1[6] = 0;
    g1[7] = 0;
    v4i z4 = {0, 0, 0, 0};
    v8i z8 = {0, 0, 0, 0, 0, 0, 0, 0};
    // clang-23 / therock-10.0 form: 6 args (g0, g1, g2, g3, g4, cpol)
    __builtin_amdgcn_tensor_load_to_lds(g0, g1, z4, z4, z8, 0);
}
#endif

// ------------------------------ WMMA GEMM ----------------------------------
// C[N_NODES x NCOL] = relu(A[N_NODES x CAT] @ W[CAT x NCOL] + bias).
// Block = 256 thr (8 waves); wave w owns 16-row stripe tileM = blk*8+w and
// computes all NCOL columns (NCOL/16 accumulators). W is consumed in KC-row
// slabs staged in LDS (TDM double-buffered when available).
template<int NCOL>
__global__ __launch_bounds__(256) void k_gemm_bias_relu(
        const float* __restrict__ A, const float* __restrict__ W,
        const float* __restrict__ bias, float* __restrict__ C) {
    constexpr int CT = NCOL / 16;        // col tiles per wave
    constexpr int KC = 32;               // K-slab rows
    constexpr int CHUNK = KC * NCOL;     // slab elements (<= 4096)
    constexpr int NCHUNK = CAT / KC;     // 12
    __shared__ float sW[2][CHUNK];

    int wave = threadIdx.x >> 5;
    int lane = threadIdx.x & 31;
    int tileM = blockIdx.x * 8 + wave;
    bool active = tileM < ROW_TILES;
    int m0 = tileM << 4;
    int half  = lane >> 4;               // lane 0-15 vs 16-31
    int ml    = lane & 15;
    int khalf = half << 1;               // {0,2}

    const float* Arow = active ? (A + (long)(m0 + ml) * CAT + khalf) : A;

    v8f acc[CT];
    #pragma unroll
    for (int ct = 0; ct < CT; ++ct) acc[ct] = (v8f){};

#if USE_TDM
    if (threadIdx.x < 32) {              // wave 0 drives the Tensor Data Mover
        tdm_load_1d(W, (unsigned int)(uintptr_t)(&sW[0][0]), CHUNK, CAT * NCOL);
    }
#endif

    for (int c = 0; c < NCHUNK; ++c) {
        int k0 = c * KC;
        int cur = c & 1;
#if USE_TDM
        if (threadIdx.x < 32) {
            if (c + 1 < NCHUNK) {        // prefetch next slab into other buffer
                tdm_load_1d(W + (long)(k0 + KC) * NCOL,
                            (unsigned int)(uintptr_t)(&sW[cur ^ 1][0]),
                            CHUNK, (CAT - k0 - KC) * NCOL);
                __builtin_amdgcn_s_wait_tensorcnt(1);   // slab c complete
            } else {
                __builtin_amdgcn_s_wait_tensorcnt(0);
            }
        }
#else
        for (int i = threadIdx.x; i < CHUNK / 4; i += 256) {
            ((float4*)&sW[cur][0])[i] = ((const float4*)(W + (long)k0 * NCOL))[i];
        }
#endif
        __syncthreads();
        if (active) {
            #pragma unroll
            for (int ks = 0; ks < KC; ks += 4) {
                const float2 av = *(const float2*)(Arow + k0 + ks);
                v2f a; a.x = av.x; a.y = av.y;
                #pragma unroll
                for (int ct = 0; ct < CT; ++ct) {
                    v2f b;                          // B frag: rows {k+khalf, +1}, col n
                    b.x = sW[cur][(ks + khalf)     * NCOL + ct * 16 + ml];
                    b.y = sW[cur][(ks + khalf + 1) * NCOL + ct * 16 + ml];
                    acc[ct] = __builtin_amdgcn_wmma_f32_16x16x4_f32(
                                  false, a, false, b, (short)0, acc[ct], false, false);
                }
            }
        }
        __syncthreads();
    }

    if (!active) return;
    // C layout: VGPR r -> rows {r, r+8} split by lane half, N = lane&15
    int rbase = m0 + (half << 3);
    #pragma unroll
    for (int ct = 0; ct < CT; ++ct) {
        int col = ct * 16 + ml;
        float bc = bias[col];
        #pragma unroll
        for (int r = 0; r < 8; ++r) {
            float v = acc[ct][r] + bc;
            C[(long)(rbase + r) * NCOL + col] = fmaxf(v, 0.0f);
        }
    }
}

// ------------------------------- readout -----------------------------------
__global__ void k_readout_max(const float* __restrict__ h2,
                              const int* __restrict__ gid,
                              float* __restrict__ out) {
    int tid = blockIdx.x * blockDim.x + threadIdx.x;     // N_NODES*16 threads
    int n = tid >> 4;
    if (n >= N_NODES) return;
    int q = (tid & 15) << 2;
    int g = gid[n];
    const float4 v = *(const float4*)(h2 + (long)n * OUT_DIM + q);
    unsigned int* op = (unsigned int*)(out + (long)g * OUT_DIM + q);
    atomicMax(op + 0, __float_as_uint(fmaxf(v.x, 0.0f)));
    atomicMax(op + 1, __float_as_uint(fmaxf(v.y, 0.0f)));
    atomicMax(op + 2, __float_as_uint(fmaxf(v.z, 0.0f)));
    atomicMax(op + 3, __float_as_uint(fmaxf(v.w, 0.0f)));
}

// ------------------------------- launcher ----------------------------------

extern "C" void kernel_launch(void* const* d_in, const int* in_sizes, int n_in,
                              void* d_out, int out_size, void* d_ws, size_t ws_size,
                              hipStream_t stream) {
    const float* x   = (const float*)d_in[0];   // 50000 x 128
    const float* W1  = (const float*)d_in[1];   // 384 x 128
    const float* b1  = (const float*)d_in[2];   // 128
    const float* W2  = (const float*)d_in[3];   // 384 x 64
    const float* b2  = (const float*)d_in[4];   // 64
    const int*   src = (const int*)d_in[5];     // 600000
    const int*   dst = (const int*)d_in[6];     // 600000
    const int*   gid = (const int*)d_in[7];     // 50000
    float* out = (float*)d_out;                 // 100 x 64

    // workspace carve-up (floats): feats | agg | deg | norm  (~103 MB)
    float* feats = (float*)d_ws;                         // N x 384
    float* agg   = feats + (long)N_NODES * CAT;          // N x 128 (also h1/h2)
    float* deg   = agg   + (long)N_NODES * 128;          // N
    float* norm  = deg   + N_NODES;                      // N

    const int B = 256;
    const long nodeVec  = (long)N_NODES * 32;
    const int  gNodeVec = (int)((nodeVec + B - 1) / B);
    const long edgeVec  = (long)N_EDGES * 32;
    const int  gEdgeVec = (int)((edgeVec + B - 1) / B);
    const int  gNodes = (N_NODES + B - 1) / B;
    const int  gEdges = (N_EDGES + B - 1) / B;
    const int  gAgg   = (N_NODES * 128 + B - 1) / B;
    const int  gGemm  = (ROW_TILES + 7) / 8;             // 391 blocks of 8 waves

    // ---- degrees & symmetric norm ----
    k_zero_f32<<<gNodes, B, 0, stream>>>(deg, N_NODES);
    k_degree<<<gEdges, B, 0, stream>>>(dst, deg, N_EDGES);
    k_norm<<<gNodes, B, 0, stream>>>(deg, norm, N_NODES);

    // ---- layer 1: feats = [x, Ax, A^2 x] ----
    k_copy128<<<gNodeVec, B, 0, stream>>>(feats, CAT, 0, x, IN_DIM, 0);
    for (int hop = 0; hop < 2; ++hop) {
        k_zero_f32<<<gAgg, B, 0, stream>>>(agg, N_NODES * 128);
        k_scatter<<<gEdgeVec, B, 0, stream>>>(feats, hop * 128, agg, norm, src, dst);
        k_scale_store<<<gNodeVec, B, 0, stream>>>(feats, (hop + 1) * 128, agg, norm);
    }
    // h1 = relu(feats @ W1 + b1) -> agg (N x 128)
    k_gemm_bias_relu<HID><<<gGemm, B, 0, stream>>>(feats, W1, b1, agg);

    // ---- layer 2: feats = [h1, A h1, A^2 h1] ----
    k_copy128<<<gNodeVec, B, 0, stream>>>(feats, CAT, 0, agg, HID, 0);
    for (int hop = 0; hop < 2; ++hop) {
        k_zero_f32<<<gAgg, B, 0, stream>>>(agg, N_NODES * 128);
        k_scatter<<<gEdgeVec, B, 0, stream>>>(feats, hop * 128, agg, norm, src, dst);
        k_scale_store<<<gNodeVec, B, 0, stream>>>(feats, (hop + 1) * 128, agg, norm);
    }
    // h2 = relu(feats @ W2 + b2) -> agg (N x 64, compact)
    k_gemm_bias_relu<OUT_DIM><<<gGemm, B, 0, stream>>>(feats, W2, b2, agg);

    // ---- per-graph max readout ----
    k_zero_f32<<<(N_GRAPHS * OUT_DIM + B - 1) / B, B, 0, stream>>>(out, N_GRAPHS * OUT_DIM);
    {
        long t = (long)N_NODES * 16;
        int g = (int)((t + B - 1) / B);
        k_readout_max<<<g, B, 0, stream>>>(agg, gid, out);
    }
}